// UniGCNIIConv_77464030151240
// MI455X (gfx1250) — compile-verified
//
#include <hip/hip_runtime.h>

// Problem constants (match reference)
#define NN  50000     // nodes
#define MM  20000     // hyperedges
#define NNZC 400000   // incidences
#define DD  256       // feature dim

typedef __attribute__((ext_vector_type(2))) float v2f;
typedef __attribute__((ext_vector_type(8))) float v8f;

// ---------------------------------------------------------------------------
// Kernel 1: gather node rows by vertex[], atomic mean-accumulate into edges.
// 64 threads per incidence (float4 per thread covers D=256). 4 incidences/block.
// ---------------------------------------------------------------------------
__global__ void k_scatter_v2e(const float* __restrict__ X,
                              const int* __restrict__ vertex,
                              const int* __restrict__ edges,
                              float* __restrict__ sums,
                              float* __restrict__ cnt) {
    int tid = threadIdx.x;
    int i   = blockIdx.x * 4 + (tid >> 6);    // incidence id
    int d4  = tid & 63;                       // float4 chunk within row
    int v = vertex[i];
    int e = edges[i];
    float4 val = reinterpret_cast<const float4*>(X + (size_t)v * DD)[d4];
    float* s = sums + (size_t)e * DD + d4 * 4;
    atomicAdd(s + 0, val.x);
    atomicAdd(s + 1, val.y);
    atomicAdd(s + 2, val.z);
    atomicAdd(s + 3, val.w);
    if (d4 == 0) atomicAdd(cnt + e, 1.0f);
}

// ---------------------------------------------------------------------------
// Kernel 2: Xe = sums / max(cnt,1) * degE   (in place over sums)
// ---------------------------------------------------------------------------
__global__ void k_finalize_edges(float* __restrict__ sums,
                                 const float* __restrict__ cnt,
                                 const float* __restrict__ degE) {
    int idx = blockIdx.x * blockDim.x + threadIdx.x;   // float4 index, M*D/4 total
    int m   = idx >> 6;                                // / (D/4)
    float scale = degE[m] / fmaxf(cnt[m], 1.0f);
    float4* p = reinterpret_cast<float4*>(sums);
    float4 t = p[idx];
    t.x *= scale; t.y *= scale; t.z *= scale; t.w *= scale;
    p[idx] = t;
}

// ---------------------------------------------------------------------------
// Kernel 3: gather edge rows by edges[], atomic sum into node accumulator Xv.
// ---------------------------------------------------------------------------
__global__ void k_scatter_e2v(const float* __restrict__ Xe,
                              const int* __restrict__ vertex,
                              const int* __restrict__ edges,
                              float* __restrict__ Xv) {
    int tid = threadIdx.x;
    int i   = blockIdx.x * 4 + (tid >> 6);
    int d4  = tid & 63;
    int v = vertex[i];
    int e = edges[i];
    float4 val = reinterpret_cast<const float4*>(Xe + (size_t)e * DD)[d4];
    float* o = Xv + (size_t)v * DD + d4 * 4;
    atomicAdd(o + 0, val.x);
    atomicAdd(o + 1, val.y);
    atomicAdd(o + 2, val.z);
    atomicAdd(o + 3, val.w);
}

// ---------------------------------------------------------------------------
// Kernel 4: one wave32 per row: scale by degV, L2-normalize (detached norm),
// mix with X0 -> Xi.  8 waves (8 rows) per 256-thread block.
// ---------------------------------------------------------------------------
__global__ void k_row_normalize(const float* __restrict__ Xv,
                                const float* __restrict__ degV,
                                const float* __restrict__ X0,
                                const float* __restrict__ alphaPtr,
                                float* __restrict__ Xi) {
    int lane = threadIdx.x & 31;
    int wave = threadIdx.x >> 5;
    int row  = blockIdx.x * 8 + wave;
    float dv    = degV[row];
    float alpha = *alphaPtr;

    const float* r  = Xv + (size_t)row * DD;
    const float* x0 = X0 + (size_t)row * DD;
    float* o        = Xi + (size_t)row * DD;

    float vals[8];
    float ss = 0.0f;
#pragma unroll
    for (int j = 0; j < 8; ++j) {
        float t = r[lane + 32 * j] * dv;
        vals[j] = t;
        ss += t * t;
    }
    // wave32 butterfly reduction
#pragma unroll
    for (int m = 16; m >= 1; m >>= 1)
        ss += __shfl_xor(ss, m, 32);

    float inv = (ss > 0.0f) ? (1.0f / sqrtf(ss)) : 0.0f;
    float oma = 1.0f - alpha;
#pragma unroll
    for (int j = 0; j < 8; ++j)
        o[lane + 32 * j] = oma * vals[j] * inv + alpha * x0[lane + 32 * j];
}

// ---------------------------------------------------------------------------
// Kernel 5: out = (1-beta)*Xi + beta*(Xi @ W) via V_WMMA_F32_16X16X4_F32.
// One wave per 16x16 output tile; K-loop of 64 WMMA ops over D=256.
// A-fragment (16x4 f32): lane<16 holds (K0,K1) of row M=lane, lane>=16 holds
// (K2,K3) of row M=lane-16  -> contiguous float2 load.
// B-fragment (4x16 f32): VGPR0 = rows K0 (lanes 0-15) / K2 (lanes 16-31),
// VGPR1 = rows K1 / K3, element N = lane&15.
// C/D (16x16 f32): elem[v] = (M = v + 8*(lane>>4), N = lane&15).
// ---------------------------------------------------------------------------
__global__ void k_gemm_wmma(const float* __restrict__ Xi,
                            const float* __restrict__ W,
                            const float* __restrict__ betaPtr,
                            float* __restrict__ out) {
    int lane = threadIdx.x & 31;
    int wave = threadIdx.x >> 5;
    int tile = blockIdx.x * 8 + wave;          // 50000 tiles total
    int row0 = (tile >> 4) << 4;               // N-dim tile (3125 of them)
    int col0 = (tile & 15) << 4;               // D-dim tile (16 of them)
    float beta = *betaPtr;

    int m    = lane & 15;
    int half = lane >> 4;                      // 0: K0/K1, 1: K2/K3
    int koff = half << 1;
    int n    = lane & 15;

    const float* arow = Xi + (size_t)(row0 + m) * DD + koff;
    const float* bcol = W  + (size_t)koff * DD + col0 + n;

    v8f acc = {};
#pragma unroll 8
    for (int k0 = 0; k0 < DD; k0 += 4) {
        float2 av = *reinterpret_cast<const float2*>(arow + k0);
        v2f a; a.x = av.x; a.y = av.y;
        v2f b;
        b.x = bcol[(size_t)k0 * DD];           // row k0+koff
        b.y = bcol[(size_t)k0 * DD + DD];      // row k0+koff+1
        acc = __builtin_amdgcn_wmma_f32_16x16x4_f32(
            /*neg_a=*/false, a, /*neg_b=*/false, b,
            /*c_mod=*/(short)0, acc, /*reuse_a=*/false, /*reuse_b=*/false);
    }

    int rbase = row0 + (half << 3);
    int c     = col0 + n;
    float omb = 1.0f - beta;
#pragma unroll
    for (int v = 0; v < 8; ++v) {
        size_t idx = (size_t)(rbase + v) * DD + c;
        out[idx] = omb * Xi[idx] + beta * acc[v];
    }
}

// ---------------------------------------------------------------------------
extern "C" void kernel_launch(void* const* d_in, const int* in_sizes, int n_in,
                              void* d_out, int out_size, void* d_ws, size_t ws_size,
                              hipStream_t stream) {
    const float* X     = (const float*)d_in[0];
    const float* X0    = (const float*)d_in[1];
    const float* W     = (const float*)d_in[2];
    const float* degV  = (const float*)d_in[3];
    const float* degE  = (const float*)d_in[4];
    const int*   vertex= (const int*)d_in[5];
    const int*   edges = (const int*)d_in[6];
    const float* alpha = (const float*)d_in[7];
    const float* beta  = (const float*)d_in[8];
    float* out = (float*)d_out;

    // Workspace layout (floats):
    float* sums = (float*)d_ws;                        // M*D  (becomes Xe)
    float* cnt  = sums + (size_t)MM * DD;              // M
    float* Xv   = cnt + MM;                            // N*D
    float* Xi   = Xv + (size_t)NN * DD;                // N*D

    // Zero accumulators (sums + cnt + Xv are contiguous).
    size_t zeroFloats = (size_t)MM * DD + MM + (size_t)NN * DD;
    hipMemsetAsync(d_ws, 0, zeroFloats * sizeof(float), stream);

    k_scatter_v2e   <<<NNZC / 4,               256, 0, stream>>>(X, vertex, edges, sums, cnt);
    k_finalize_edges<<<(MM * DD / 4) / 256,    256, 0, stream>>>(sums, cnt, degE);
    k_scatter_e2v   <<<NNZC / 4,               256, 0, stream>>>(sums, vertex, edges, Xv);
    k_row_normalize <<<NN / 8,                 256, 0, stream>>>(Xv, degV, X0, alpha, Xi);
    k_gemm_wmma     <<<(NN / 16) * (DD / 16) / 8, 256, 0, stream>>>(Xi, W, beta, out);
}